// GraphModel_52209622450442
// MI455X (gfx1250) — compile-verified
//
#include <hip/hip_runtime.h>
#include <math.h>

// Problem constants (fixed by the reference)
#define NN      10000
#define FIN     128
#define FHID    64
#define FOUT    32
#define EE      320000
#define NTILE   625          // NN / 16
#define MAX_LOGSTD 10.0f

typedef float v2f __attribute__((ext_vector_type(2)));
typedef float v8f __attribute__((ext_vector_type(8)));

// ---------------- degree / dinv ----------------
__global__ void deg_kernel(const int* __restrict__ ei, float* __restrict__ deg) {
    int t = blockIdx.x * blockDim.x + threadIdx.x;
    if (t < EE) atomicAdd(&deg[ei[EE + t]], 1.0f);     // dst = row 1
}

__global__ void dinv_kernel(float* __restrict__ deg) {
    int t = blockIdx.x * blockDim.x + threadIdx.x;
    if (t < NN) deg[t] = rsqrtf(deg[t] + 1.0f);        // +1 self loop; always > 0
}

// ---------------- dense transform H = X @ W ----------------
__global__ void matmul_kernel(const float* __restrict__ X, const float* __restrict__ W,
                              float* __restrict__ H, int K, int Fo) {
    int t = blockIdx.x * blockDim.x + threadIdx.x;
    if (t >= NN * Fo) return;
    int i = t / Fo, f = t - i * Fo;
    const float* xr = X + (size_t)i * K;
    float acc = 0.0f;
    for (int k = 0; k < K; ++k) acc = fmaf(xr[k], W[(size_t)k * Fo + f], acc);
    H[t] = acc;
}

// ---------------- edge scatter: AG[dst] += dinv[s]*dinv[d]*H[src] ----------------
__global__ void scatter_kernel(const int* __restrict__ ei, const float* __restrict__ dinv,
                               const float* __restrict__ H, float* __restrict__ AG, int F) {
    long long t = (long long)blockIdx.x * blockDim.x + threadIdx.x;
    if (t >= (long long)EE * F) return;
    int e = (int)(t / F), f = (int)(t - (long long)e * F);
    int s = ei[e], d = ei[EE + e];
    float nrm = dinv[s] * dinv[d];
    atomicAdd(&AG[(size_t)d * F + f], nrm * H[(size_t)s * F + f]);
}

// ---------------- combine: out = [relu](agg + dinv^2*h + b) ----------------
__global__ void combine_kernel(const float* __restrict__ AG, const float* __restrict__ H,
                               const float* __restrict__ dinv, const float* __restrict__ b,
                               float* __restrict__ Out, int F, int do_relu) {
    int t = blockIdx.x * blockDim.x + threadIdx.x;
    if (t >= NN * F) return;
    int i = t / F, f = t - i * F;
    float di = dinv[i];
    float v = AG[t] + di * di * H[t] + b[f];
    Out[t] = do_relu ? fmaxf(v, 0.0f) : v;
}

// ---------------- reparameterize: z = mu + eps*exp(min(logstd, 10)) ----------------
__global__ void z_kernel(const float* __restrict__ AGmu, const float* __restrict__ Hmu,
                         const float* __restrict__ bmu,
                         const float* __restrict__ AGls, const float* __restrict__ Hls,
                         const float* __restrict__ bls,
                         const float* __restrict__ dinv, const float* __restrict__ eps,
                         float* __restrict__ Z) {
    int t = blockIdx.x * blockDim.x + threadIdx.x;
    if (t >= NN * FOUT) return;
    int i = t / FOUT, f = t - i * FOUT;
    float d2 = dinv[i] * dinv[i];
    float mu = AGmu[t] + d2 * Hmu[t] + bmu[f];
    float ls = AGls[t] + d2 * Hls[t] + bls[f];
    ls = fminf(ls, MAX_LOGSTD);
    Z[t] = fmaf(eps[t], expf(ls), mu);
}

// ---------------- decoder: out = sigmoid(Z @ Z^T) via fp32 WMMA 16x16x4 ----------------
// One wave per 16x16 output tile. K=32 -> 8 WMMA steps.
// A-frag (16x4 f32): lane l: m = l&15; lanes 0-15 hold K={0,1}, lanes 16-31 hold K={2,3}.
// B-frag (4x16, B = Z^T): same per-lane indexing with n = l&15 (symmetric!).
// C/D (16x16 f32, 8 VGPRs): lane l, vgpr r -> row = r + 8*(l>>4), col = l&15.
__global__ __launch_bounds__(256) void decoder_kernel(const float* __restrict__ Z,
                                                      float* __restrict__ out) {
    const int wave = threadIdx.x >> 5;            // 8 waves / block
    const int lane = threadIdx.x & 31;
    const int tn = blockIdx.x * 8 + wave;         // tile col
    const int tm = blockIdx.y;                    // tile row
    if (tn >= NTILE) return;                      // wave-uniform: EXEC stays all-ones

    const int r  = lane & 15;
    const int hi = lane >> 4;
    const float* za = Z + (size_t)(tm * 16 + r) * FOUT + 2 * hi;
    const float* zb = Z + (size_t)(tn * 16 + r) * FOUT + 2 * hi;

    v8f c = {};
#pragma unroll
    for (int s = 0; s < 8; ++s) {
        v2f a, b;
        a.x = za[4 * s]; a.y = za[4 * s + 1];
        b.x = zb[4 * s]; b.y = zb[4 * s + 1];
        c = __builtin_amdgcn_wmma_f32_16x16x4_f32(false, a, false, b,
                                                  (short)0, c, false, false);
    }

    const int col = tn * 16 + r;
#pragma unroll
    for (int rr = 0; rr < 8; ++rr) {
        int row = tm * 16 + rr + 8 * hi;
        // fast sigmoid: single v_exp_f32 + single v_rcp_f32 (avoid IEEE div sequence)
        float e = __expf(-c[rr]);
        out[(size_t)row * NN + col] = __builtin_amdgcn_rcpf(1.0f + e);
    }
}

// ---------------- driver ----------------
extern "C" void kernel_launch(void* const* d_in, const int* in_sizes, int n_in,
                              void* d_out, int out_size, void* d_ws, size_t ws_size,
                              hipStream_t stream) {
    (void)in_sizes; (void)n_in; (void)out_size; (void)ws_size;

    const float* x   = (const float*)d_in[0];
    const int*   ei  = (const int*)d_in[1];
    const float* eps = (const float*)d_in[2];
    const float* W0  = (const float*)d_in[3];
    const float* b0  = (const float*)d_in[4];
    const float* W1  = (const float*)d_in[5];
    const float* b1  = (const float*)d_in[6];
    const float* Wmu = (const float*)d_in[7];
    const float* bmu = (const float*)d_in[8];
    const float* Wls = (const float*)d_in[9];
    const float* bls = (const float*)d_in[10];
    float* out = (float*)d_out;

    // workspace layout (floats): total ~3.53M floats (~14.1 MB)
    float* w    = (float*)d_ws;
    float* dinv = w;                      // NN
    float* A    = dinv + NN;              // NN*FHID  (aggregation)
    float* B    = A + (size_t)NN * FHID;  // NN*FHID  (dense transform)
    float* C    = B + (size_t)NN * FHID;  // NN*FHID  (layer activations)
    float* Amu  = C + (size_t)NN * FHID;  // NN*FOUT
    float* Bmu  = Amu + (size_t)NN * FOUT;
    float* Als  = Bmu + (size_t)NN * FOUT;
    float* Bls  = Als + (size_t)NN * FOUT;
    float* Zb   = Bls + (size_t)NN * FOUT;

    const int T = 256;
    auto blocks = [](long long n, int t) { return (unsigned)((n + t - 1) / t); };

    // gcn_norm
    hipMemsetAsync(dinv, 0, NN * sizeof(float), stream);
    deg_kernel<<<blocks(EE, T), T, 0, stream>>>(ei, dinv);
    dinv_kernel<<<blocks(NN, T), T, 0, stream>>>(dinv);

    // conv0: C = relu(gcn(x, W0, b0))
    matmul_kernel<<<blocks((long long)NN * FHID, T), T, 0, stream>>>(x, W0, B, FIN, FHID);
    hipMemsetAsync(A, 0, (size_t)NN * FHID * sizeof(float), stream);
    scatter_kernel<<<blocks((long long)EE * FHID, T), T, 0, stream>>>(ei, dinv, B, A, FHID);
    combine_kernel<<<blocks((long long)NN * FHID, T), T, 0, stream>>>(A, B, dinv, b0, C, FHID, 1);

    // conv1: C = relu(gcn(C, W1, b1))
    matmul_kernel<<<blocks((long long)NN * FHID, T), T, 0, stream>>>(C, W1, B, FHID, FHID);
    hipMemsetAsync(A, 0, (size_t)NN * FHID * sizeof(float), stream);
    scatter_kernel<<<blocks((long long)EE * FHID, T), T, 0, stream>>>(ei, dinv, B, A, FHID);
    combine_kernel<<<blocks((long long)NN * FHID, T), T, 0, stream>>>(A, B, dinv, b1, C, FHID, 1);

    // mu head
    matmul_kernel<<<blocks((long long)NN * FOUT, T), T, 0, stream>>>(C, Wmu, Bmu, FHID, FOUT);
    hipMemsetAsync(Amu, 0, (size_t)NN * FOUT * sizeof(float), stream);
    scatter_kernel<<<blocks((long long)EE * FOUT, T), T, 0, stream>>>(ei, dinv, Bmu, Amu, FOUT);

    // logstd head
    matmul_kernel<<<blocks((long long)NN * FOUT, T), T, 0, stream>>>(C, Wls, Bls, FHID, FOUT);
    hipMemsetAsync(Als, 0, (size_t)NN * FOUT * sizeof(float), stream);
    scatter_kernel<<<blocks((long long)EE * FOUT, T), T, 0, stream>>>(ei, dinv, Bls, Als, FOUT);

    // z = mu + eps * exp(min(logstd, 10))
    z_kernel<<<blocks((long long)NN * FOUT, T), T, 0, stream>>>(Amu, Bmu, bmu, Als, Bls, bls,
                                                                dinv, eps, Zb);

    // adj = sigmoid(z z^T) : 625x625 16x16 tiles, 8 waves (tiles) per block
    dim3 grid((NTILE + 7) / 8, NTILE);
    decoder_kernel<<<grid, 256, 0, stream>>>(Zb, out);
}